// EnvironmentLight_13623636263031
// MI455X (gfx1250) — compile-verified
//
#include <hip/hip_runtime.h>
#include <cstdint>

#define TPB 256
#define MIN_ROUGHNESS 0.08f
#define MAX_ROUGHNESS 0.5f

// 12-byte texel (float3 with 4-byte alignment -> global_load_b96)
struct __attribute__((aligned(4))) f3 { float x, y, z; };
// 8-byte LUT texel -> global_load_b64
struct __attribute__((aligned(8))) f2 { float x, y; };

__device__ __forceinline__ float clampf(float v, float lo, float hi) {
    return fminf(fmaxf(v, lo), hi);
}

// linear -> sRGB, pow via v_log_f32 / v_exp_f32
__device__ __forceinline__ float l2s(float x) {
    float xs = (x > 0.0031308f) ? x : 1.0f;
    float s  = 1.055f * __builtin_amdgcn_exp2f(__builtin_amdgcn_logf(xs) * (1.0f / 2.4f)) - 0.055f;
    return (x > 0.0031308f) ? s : 12.92f * x;
}

__device__ __forceinline__ void cube_face_uv(float x, float y, float z,
                                             int& face, float& u, float& v) {
    float ax = fabsf(x), ay = fabsf(y), az = fabsf(z);
    bool is_x = (ax >= ay) && (ax >= az);
    bool is_y = (!is_x) && (ay >= az);
    face = is_x ? ((x >= 0.f) ? 0 : 1)
                : (is_y ? ((y >= 0.f) ? 2 : 3)
                        : ((z >= 0.f) ? 4 : 5));
    float ma = fmaxf(is_x ? ax : (is_y ? ay : az), 1e-20f);
    float un = (face == 0) ? -z : ((face == 1) ? z : ((face == 5) ? -x : x));
    float vn = (face == 2) ?  z : ((face == 3) ? -z : -y);
    u = un / ma;
    v = vn / ma;
}

// weighted bilinear cubemap tap, accumulate into (ar,ag,ab).
// Works for both global (vmem) and LDS (ds) texel arrays.
__device__ __forceinline__ void bilin_cube_acc(const f3* t, int R,
                                               int face, float u, float v, float w,
                                               float& ar, float& ag, float& ab) {
    float Rf = (float)R;
    float tu = (u * 0.5f + 0.5f) * Rf - 0.5f;
    float tv = (v * 0.5f + 0.5f) * Rf - 0.5f;
    float x0 = floorf(tu), y0 = floorf(tv);
    float fx = tu - x0,    fy = tv - y0;
    float Rm1 = Rf - 1.0f;
    int x0i = (int)clampf(x0,       0.f, Rm1);
    int x1i = (int)clampf(x0 + 1.f, 0.f, Rm1);
    int y0i = (int)clampf(y0,       0.f, Rm1);
    int y1i = (int)clampf(y0 + 1.f, 0.f, Rm1);
    int b = face * R * R;
    f3 c00 = t[b + y0i * R + x0i];
    f3 c01 = t[b + y0i * R + x1i];
    f3 c10 = t[b + y1i * R + x0i];
    f3 c11 = t[b + y1i * R + x1i];
    float gx = 1.f - fx, gy = 1.f - fy;
    ar += w * ((c00.x * gx + c01.x * fx) * gy + (c10.x * gx + c11.x * fx) * fy);
    ag += w * ((c00.y * gx + c01.y * fx) * gy + (c10.y * gx + c11.y * fx) * fy);
    ab += w * ((c00.z * gx + c01.z * fx) * gy + (c10.z * gx + c11.z * fx) * fy);
}

// select one of six uniform pointers by a divergent level index (cndmask chain,
// avoids scratch from dynamically indexing a local pointer array)
__device__ __forceinline__ const float* sel6(int l,
    const float* a0, const float* a1, const float* a2,
    const float* a3, const float* a4, const float* a5) {
    const float* p = a0;
    p = (l == 1) ? a1 : p;
    p = (l == 2) ? a2 : p;
    p = (l == 3) ? a3 : p;
    p = (l == 4) ? a4 : p;
    p = (l == 5) ? a5 : p;
    return p;
}

__global__ __launch_bounds__(TPB) void envlight_kernel(
    const float* __restrict__ gb_pos,
    const float* __restrict__ gb_normal,
    const float* __restrict__ basecolor,
    const float* __restrict__ metallic,
    const float* __restrict__ roughness,
    const float* __restrict__ view_pos,
    const float* __restrict__ diffuse,
    const float* __restrict__ fg_lut,
    const float* __restrict__ sp0, const float* __restrict__ sp1,
    const float* __restrict__ sp2, const float* __restrict__ sp3,
    const float* __restrict__ sp4, const float* __restrict__ sp5,
    float* __restrict__ out, int npix)
{
    // 6*16*16*3 floats = 18432 bytes = 2304 x b64 chunks = 9 per thread (exact)
    __shared__ float s_diff[6 * 16 * 16 * 3];

    // keep a visible (never-executed) write so the compiler can't assume the
    // LDS array is only read; real writes come from the async-DMA asm below.
    if (npix == -1) s_diff[0] = 0.f;

    // ---- CDNA5 async global->LDS staging of the diffuse cubemap ----
    {
        unsigned tid      = threadIdx.x;
        // low 32 bits of an LDS-aperture flat address == wave-relative LDS offset
        unsigned lds_base = (unsigned)(uintptr_t)(&s_diff[0]);
        unsigned long long gbase = (unsigned long long)(uintptr_t)diffuse;
        #pragma unroll
        for (int i = 0; i < 9; ++i) {
            unsigned goff  = (tid + (unsigned)i * TPB) * 8u;
            unsigned laddr = lds_base + goff;
            asm volatile("global_load_async_to_lds_b64 %0, %1, %2"
                         :: "v"(laddr), "v"(goff), "s"(gbase) : "memory");
        }
        asm volatile("s_wait_asynccnt 0" ::: "memory");
    }
    __syncthreads();

    int p = blockIdx.x * TPB + threadIdx.x;
    if (p >= npix) return;

    f3 pos = *(const f3*)(gb_pos    + 3 * (size_t)p);
    f3 nrm = *(const f3*)(gb_normal + 3 * (size_t)p);
    f3 bc  = *(const f3*)(basecolor + 3 * (size_t)p);
    float met = metallic[p];
    float rgh = roughness[p];
    float vpx = view_pos[0], vpy = view_pos[1], vpz = view_pos[2];

    // wo = safe_normalize(view_pos - gb_pos)
    float wx = vpx - pos.x, wy = vpy - pos.y, wz = vpz - pos.z;
    float inv = rsqrtf(fmaxf(wx * wx + wy * wy + wz * wz, 1e-20f));
    wx *= inv; wy *= inv; wz *= inv;

    float one_m = 1.f - met;
    float dax = one_m * bc.x, day = one_m * bc.y, daz = one_m * bc.z;
    float sax = 0.04f * one_m + met * bc.x;
    float say = 0.04f * one_m + met * bc.y;
    float saz = 0.04f * one_m + met * bc.z;

    float ndv_raw = wx * nrm.x + wy * nrm.y + wz * nrm.z;

    // reflvec = safe_normalize(2*dot(wo,n)*n - wo)
    float rx = 2.f * ndv_raw * nrm.x - wx;
    float ry = 2.f * ndv_raw * nrm.y - wy;
    float rz = 2.f * ndv_raw * nrm.z - wz;
    float rinv = rsqrtf(fmaxf(rx * rx + ry * ry + rz * rz, 1e-20f));
    rx *= rinv; ry *= rinv; rz *= rinv;

    // ambient = sample diffuse cubemap (from LDS) along the normal
    int fN; float uN, vN;
    cube_face_uv(nrm.x, nrm.y, nrm.z, fN, uN, vN);
    float amr = 0.f, amg = 0.f, amb = 0.f;
    bilin_cube_acc((const f3*)s_diff, 16, fN, uN, vN, 1.f, amr, amg, amb);
    float dlr = amr * dax, dlg = amg * day, dlb = amb * daz;

    // FG LUT bilinear (clamped), uv = (NdotV, roughness), 256x256x2
    float ndv = fmaxf(ndv_raw, 0.0001f);
    {
        // fallthrough scope to keep names tidy
    }
    float tu = ndv * 256.f - 0.5f;
    float tv = rgh * 256.f - 0.5f;
    float fx0 = floorf(tu), fy0 = floorf(tv);
    float ffx = tu - fx0,   ffy = tv - fy0;
    int lx0 = (int)clampf(fx0,       0.f, 255.f);
    int lx1 = (int)clampf(fx0 + 1.f, 0.f, 255.f);
    int ly0 = (int)clampf(fy0,       0.f, 255.f);
    int ly1 = (int)clampf(fy0 + 1.f, 0.f, 255.f);
    const f2* lut = (const f2*)fg_lut;
    f2 l00 = lut[ly0 * 256 + lx0];
    f2 l01 = lut[ly0 * 256 + lx1];
    f2 l10 = lut[ly1 * 256 + lx0];
    f2 l11 = lut[ly1 * 256 + lx1];
    float gfx = 1.f - ffx, gfy = 1.f - ffy;
    float fgx = (l00.x * gfx + l01.x * ffx) * gfy + (l10.x * gfx + l11.x * ffx) * ffy;
    float fgy = (l00.y * gfx + l01.y * ffx) * gfy + (l10.y * gfx + l11.y * ffx) * ffy;

    // mip = clip(get_mip(roughness), 0, 5)
    float lo = (clampf(rgh, MIN_ROUGHNESS, MAX_ROUGHNESS) - MIN_ROUGHNESS)
               / (MAX_ROUGHNESS - MIN_ROUGHNESS) * 4.0f;
    float hi = (clampf(rgh, MAX_ROUGHNESS, 1.0f) - MAX_ROUGHNESS)
               / (1.0f - MAX_ROUGHNESS) + 4.0f;
    float mip = clampf((rgh < MAX_ROUGHNESS) ? lo : hi, 0.f, 5.f);

    // only two adjacent mips carry nonzero weight; all other terms of the
    // reference's 6-term sum are exactly 0 and adding +0 is a no-op in fp.
    int   l0 = (int)floorf(mip); l0 = (l0 < 5) ? l0 : 5;
    int   l1 = (l0 + 1 < 5) ? (l0 + 1) : 5;
    float w1 = mip - (float)l0;
    float w0 = 1.f - w1;

    int fR; float uR, vR;
    cube_face_uv(rx, ry, rz, fR, uR, vR);

    const float* t0 = sel6(l0, sp0, sp1, sp2, sp3, sp4, sp5);
    const float* t1 = sel6(l1, sp0, sp1, sp2, sp3, sp4, sp5);
    int R0 = 512 >> l0;
    int R1 = 512 >> l1;

    float spr = 0.f, spg = 0.f, spb = 0.f;
    bilin_cube_acc((const f3*)t0, R0, fR, uR, vR, w0, spr, spg, spb);
    bilin_cube_acc((const f3*)t1, R1, fR, uR, vR, w1, spr, spg, spb);

    float rr = sax * fgx + fgy;
    float rg = say * fgx + fgy;
    float rb = saz * fgx + fgy;

    float cr = clampf(spr * rr + dlr, 0.f, 1.f);
    float cg = clampf(spg * rg + dlg, 0.f, 1.f);
    float cb = clampf(spb * rb + dlb, 0.f, 1.f);

    f3 res; res.x = l2s(cr); res.y = l2s(cg); res.z = l2s(cb);
    ((f3*)out)[p] = res;
}

extern "C" void kernel_launch(void* const* d_in, const int* in_sizes, int n_in,
                              void* d_out, int out_size, void* d_ws, size_t ws_size,
                              hipStream_t stream) {
    const float* gb_pos    = (const float*)d_in[0];
    const float* gb_normal = (const float*)d_in[1];
    const float* basecolor = (const float*)d_in[2];
    const float* metallic  = (const float*)d_in[3];
    const float* roughness = (const float*)d_in[4];
    const float* view_pos  = (const float*)d_in[5];
    const float* diffuse   = (const float*)d_in[6];
    const float* fg_lut    = (const float*)d_in[7];
    const float* s0 = (const float*)d_in[8];
    const float* s1 = (const float*)d_in[9];
    const float* s2 = (const float*)d_in[10];
    const float* s3 = (const float*)d_in[11];
    const float* s4 = (const float*)d_in[12];
    const float* s5 = (const float*)d_in[13];

    int npix   = in_sizes[0] / 3;               // (1,1024,1024,3) -> 1M pixels
    int blocks = (npix + TPB - 1) / TPB;

    envlight_kernel<<<blocks, TPB, 0, stream>>>(
        gb_pos, gb_normal, basecolor, metallic, roughness, view_pos,
        diffuse, fg_lut, s0, s1, s2, s3, s4, s5,
        (float*)d_out, npix);
}